// HIST_loss_55018531061908
// MI455X (gfx1250) — compile-verified
//
#include <hip/hip_runtime.h>
#include <math.h>

typedef float v2f __attribute__((ext_vector_type(2)));
typedef float v8f __attribute__((ext_vector_type(8)));

static constexpr int IH = 512, IW = 512;
static constexpr int BINS = 25;
static constexpr int NPLANES = 192;            // 96 x-planes then 96 y-planes
static constexpr int PLANE_ELEMS = IH * IW;
static constexpr int TILES_PER_PLANE = 256;    // 16x16 grid of 16x16 output tiles
static constexpr int WAVES_PER_BLOCK = 4;
static constexpr int PR = 37;                  // live patch rows/cols
static constexpr int PS = 40;                  // padded LDS patch row stride (floats)
static constexpr int TS = 16;                  // tbuf row stride (floats)
static constexpr int TROWS = 40;               // padded tbuf rows (rows 37..39 zero)

// ---------------------------------------------------------------------------
// Kernel 1: zero the per-plane histogram workspace (d_ws is poisoned 0xAA).
// ---------------------------------------------------------------------------
__global__ void zero_hist_kernel(float* __restrict__ g, int n) {
    int i = blockIdx.x * blockDim.x + threadIdx.x;
    if (i < n) g[i] = 0.0f;
}

// ---------------------------------------------------------------------------
// Kernel 2: fused blur_pool2d (7x7 pascal, stride 2, zero pad 3) + 25-bin
// per-plane histogram. One wave == one 16x16 output tile of one plane.
//   - patch 37x(37..40) global -> LDS; interior tiles take an unguarded,
//     fully-unrolled path (base + immediate offsets only)
//   - horizontal 7-tap conv (stride 2) via VALU (float2 LDS loads) -> tbuf
//   - vertical 7-tap conv (stride 2) as banded matmul:
//       O(16x16) = Wv(16x40) * T(40x16)  via 10x V_WMMA_F32_16X16X4_F32
//     A-matrix chunks come from a per-block LDS table (one ds_load_b64 each),
//     B-matrix loads are unconditional thanks to the zero-padded tbuf rows.
//   - histogram D-matrix values (8 per lane) branch-free via LDS atomics
//     (increment = valid ? 1 : 0), then flush to global with f32 atomics
// ---------------------------------------------------------------------------
__global__ __launch_bounds__(128) void blur_hist_kernel(
        const float* __restrict__ x, const float* __restrict__ y,
        float* __restrict__ gHist) {
    __shared__ float        awtab[10 * 64];                 // A chunks, per-lane pairs
    __shared__ float        patch[WAVES_PER_BLOCK][PR * PS];
    __shared__ float        tbuf [WAVES_PER_BLOCK][TROWS * TS];
    __shared__ unsigned int hist [WAVES_PER_BLOCK][32];

    const int lane = threadIdx.x & 31;
    const int wave = threadIdx.x >> 5;

    const float w0 = 1.f/64.f, w1 = 6.f/64.f, w2 = 15.f/64.f, w3 = 20.f/64.f;

    // --- one-time per-block: banded A-matrix table  Wv[m][k] = w[k-2m]
    // layout matches 32-bit A 16x4 VGPR striping: entry (j, lane, half):
    //   m = lane&15, k = 4j + ((lane>>4)<<1) + half
    for (int e = threadIdx.x; e < 640; e += 128) {
        const int j    = e >> 6;
        const int r    = e & 63;
        const int ln   = r >> 1;
        const int half = r & 1;
        const int lm   = ln & 15;
        const int k    = 4 * j + (((ln >> 4) << 1) | half);
        const int d    = k - 2 * lm;
        float v = 0.0f;
        if (d == 3)                v = w3;
        else if (d == 2 || d == 4) v = w2;
        else if (d == 1 || d == 5) v = w1;
        else if (d == 0 || d == 6) v = w0;
        awtab[e] = v;
    }
    // per-wave: zero hist bins and tbuf pad rows (37..39)
    if (lane < BINS) hist[wave][lane] = 0u;
    for (int e = lane; e < 3 * TS; e += 32) tbuf[wave][PR * TS + e] = 0.0f;

    const int gid   = blockIdx.x * WAVES_PER_BLOCK + wave;  // 0 .. 192*256-1
    const int plane = gid >> 8;
    const int tile  = gid & 255;
    const int tr    = tile >> 4;
    const int tc    = tile & 15;

    const float* src = (plane < 96) ? (x + (size_t)plane * PLANE_ELEMS)
                                    : (y + (size_t)(plane - 96) * PLANE_ELEMS);
    const int gr0 = tr * 32 - 3;   // input-row origin of patch (includes pad)
    const int gc0 = tc * 32 - 3;   // input-col origin of patch

    // --- stage global -> LDS
    const bool interior = (tr > 0) & (tr < 15) & (tc > 0) & (tc < 15);
    if (interior) {
        // all addresses in-bounds: rows gr0..gr0+36 <= 481, cols gc0..gc0+39 <= 484
        const float* g = src + (size_t)gr0 * IW + gc0;
        float* pd = &patch[wave][0];
        {
            const float* gl = g + lane;
            float* dl = pd + lane;
            #pragma unroll
            for (int i = 0; i < PR; ++i) dl[i * PS] = gl[i * IW];
        }
        if (lane < 8) {                     // cols 32..39 (37..39 are dead pad)
            const float* gl = g + 32 + lane;
            float* dl = pd + 32 + lane;
            #pragma unroll
            for (int i = 0; i < PR; ++i) dl[i * PS] = gl[i * IW];
        }
    } else {
        for (int i = 0; i < PR; ++i) {
            const int  r   = gr0 + i;
            const bool rin = (r >= 0) && (r < IH);
            for (int cb = 0; cb < 64; cb += 32) {
                const int c = cb + lane;
                if (c < PR) {
                    const int gc = gc0 + c;
                    float v = 0.0f;
                    if (rin && gc >= 0 && gc < IW) v = src[(size_t)r * IW + gc];
                    patch[wave][i * PS + c] = v;
                }
            }
        }
    }
    __syncthreads();

    // --- stage 1: horizontal conv, stride 2: T[i][n] = sum_t w[t]*P[i][2n+t]
    // per lane: n fixed, rows i = (lane>>4) + 2k  -> base + immediate offsets
    {
        const int n  = lane & 15;
        const int i0 = lane >> 4;
        const float* prow = &patch[wave][2 * n];   // 8B-aligned base
        float* trow = &tbuf[wave][n];
        for (int i = i0; i < PR; i += 2) {
            const v2f* p2 = (const v2f*)&prow[i * PS];
            const v2f q0 = p2[0], q1 = p2[1], q2 = p2[2];
            const float p6 = prow[i * PS + 6];
            trow[i * TS] = q0.x*w0 + q0.y*w1 + q1.x*w2 + q1.y*w3 +
                           q2.x*w2 + q2.y*w1 + p6*w0;
        }
    }
    __syncthreads();

    // --- stage 2: vertical conv as WMMA chain, O = Wv * T (all loads unguarded)
    v8f acc = {};
    const int khalf = (lane >> 4) << 1;                    // 0 or 2
    const float* at = &awtab[lane << 1];                   // + j*64 floats
    const float* tb = &tbuf[wave][khalf * TS + (lane & 15)];
    #pragma unroll
    for (int j = 0; j < 10; ++j) {
        const v2f a = *(const v2f*)&at[j * 64];
        v2f b;
        b.x = tb[j * 64];
        b.y = tb[j * 64 + 16];
        acc = __builtin_amdgcn_wmma_f32_16x16x4_f32(
                  false, a, false, b, (short)0, acc, false, false);
    }

    // --- branch-free histogram of the 8 output pixels this lane owns
    // torch.histc semantics: count only v in [0,1]; v==1 lands in last bin.
    // Always issue the LDS atomic; invalid values add 0 to a clamped bin.
    unsigned int* hrow = &hist[wave][0];
    #pragma unroll
    for (int vi = 0; vi < 8; ++vi) {
        const float v = acc[vi];
        int b = (int)(v * 25.0f);
        b = (b < 0) ? 0 : ((b > 24) ? 24 : b);
        const unsigned int inc = (v >= 0.0f && v <= 1.0f) ? 1u : 0u;
        atomicAdd(&hrow[b], inc);
    }
    __syncthreads();

    if (lane < BINS) {
        atomicAdd(&gHist[plane * BINS + lane], (float)hist[wave][lane]);
    }
}

// ---------------------------------------------------------------------------
// Kernel 3: per-(b,c) cosine similarity of normalized histograms, mean -> out
// ---------------------------------------------------------------------------
__global__ void cosine_kernel(const float* __restrict__ gHist,
                              float* __restrict__ out) {
    __shared__ float red[128];
    const int t = threadIdx.x;
    float cosv = 0.0f;
    if (t < 96) {
        const float inv = 1.0f / (float)(IH * IW);   // /hw normalization
        const float* xh = gHist + t * BINS;
        const float* yh = gHist + (96 + t) * BINS;
        float dot = 0.f, nx = 0.f, ny = 0.f;
        #pragma unroll
        for (int i = 0; i < BINS; ++i) {
            const float a = xh[i] * inv;
            const float b = yh[i] * inv;
            dot += a * b; nx += a * a; ny += b * b;
        }
        nx = fmaxf(sqrtf(nx), 1e-6f);
        ny = fmaxf(sqrtf(ny), 1e-6f);
        cosv = dot / (nx * ny);
    }
    red[t] = cosv;
    __syncthreads();
    for (int s = 64; s > 0; s >>= 1) {
        if (t < s) red[t] += red[t + s];
        __syncthreads();
    }
    if (t == 0) out[0] = red[0] / 96.0f;
}

extern "C" void kernel_launch(void* const* d_in, const int* in_sizes, int n_in,
                              void* d_out, int out_size, void* d_ws, size_t ws_size,
                              hipStream_t stream) {
    const float* x = (const float*)d_in[0];
    const float* y = (const float*)d_in[1];
    float* gHist = (float*)d_ws;                 // 192*25 floats = 19.2 KB

    const int nh = NPLANES * BINS;
    zero_hist_kernel<<<(nh + 255) / 256, 256, 0, stream>>>(gHist, nh);

    const int nblocks = NPLANES * TILES_PER_PLANE / WAVES_PER_BLOCK;  // 12288
    blur_hist_kernel<<<nblocks, WAVES_PER_BLOCK * 32, 0, stream>>>(x, y, gHist);

    cosine_kernel<<<1, 128, 0, stream>>>(gHist, (float*)d_out);
}